// Pool_35089882808587
// MI455X (gfx1250) — compile-verified
//
#include <hip/hip_runtime.h>
#include <hip/hip_bf16.h>

typedef float v2f __attribute__((ext_vector_type(2)));
typedef float v8f __attribute__((ext_vector_type(8)));

#define NKER 36      // rf mask kernels
#define HW   64      // input H = W
#define OW   32      // output H = W (after 2x2 sum-pool)
#define RUNS 256     // 16 local-m rows * 16 local-n cols staged per block
#define RSTR 37      // LDS run stride (dwords): conflict-free, runs don't overlap

// One workgroup = 8 waves = 256 threads.
// Block owns (m-tile of 16 (b,c) rows, one output row oh, n-tile of 16 kernels).
// Wave w computes ow = 4w..4w+3, one v_wmma_f32_16x16x4_f32 per ow
// (K=4 == the 2x2 pooling window), stages D in LDS, then the whole block
// streams the 16x16x32 result out with coalesced float4 stores.
__global__ __launch_bounds__(256) void rfpool_wmma_16x16x4(
    const float* __restrict__ u,    // (8,32,64,64) -> (256, 64, 64)
    const float* __restrict__ rfs,  // (36,64,64)
    float* __restrict__ out)        // (256, 36, 32, 32)
{
    __shared__ float lds[RUNS * RSTR + 2];

    const int tid  = threadIdx.x;
    const int lane = tid & 31;
    const int wave = tid >> 5;
    const int lrow = lane & 15;   // M row (A) / N col (B) within tile
    const int hi   = lane >> 4;   // selects K pair {0,1} vs {2,3} == kh

    int bid = blockIdx.x;
    const int nt = bid % 3;  bid /= 3;   // n-tile: k0 = 16*nt
    const int oh = bid % OW; bid /= OW;  // output row
    const int mt = bid;                  // m-tile: m0 = 16*mt  (0..15)

    // Clamp k for the ragged last n-tile (k=36..47): produces garbage D
    // columns that are never written (WMMA columns are independent in B).
    const int krow = min(nt * 16 + lrow, NKER - 1);

    // Per-lane row bases: input row h = 2*oh + hi  (hi == kh)
    const float* ua = u   + ((mt * 16 + lrow) * (HW * HW) + (2 * oh + hi) * HW);
    const float* ba = rfs + (krow * (HW * HW) + (2 * oh + hi) * HW);

#pragma unroll
    for (int i = 0; i < 4; ++i) {
        const int ow = wave * 4 + i;
        // float2 load == A VGPR0 (K=2*hi+0), VGPR1 (K=2*hi+1): ISA 16x4 f32 A layout.
        v2f a = *(const v2f*)(ua + 2 * ow);
        // B mirrors A: VGPR0 = row K=2*hi, VGPR1 = row K=2*hi+1, N = lane%16.
        v2f b = *(const v2f*)(ba + 2 * ow);
        v8f c = {};
        // D[m,k] = sum_{j=0..3} U[m,j] * R[j,k]   (the masked 2x2 sum-pool)
        v8f d = __builtin_amdgcn_wmma_f32_16x16x4_f32(
            /*neg_a=*/false, a, /*neg_b=*/false, b,
            /*c_mod=*/(short)0, c, /*reuse_a=*/false, /*reuse_b=*/false);

        // Stage: D layout is lane -> N=lane%16, VGPR v -> M = v + 8*(lane/16).
#pragma unroll
        for (int v = 0; v < 8; ++v) {
            const int lm  = v + 8 * hi;
            const int run = lm * 16 + lrow;
            lds[run * RSTR + hi + ow] = d[v];  // +hi keeps both half-waves conflict-free
        }
    }

    __syncthreads();

    // Coalesced write-out: 256 runs of 32 floats; 8 lanes x float4 = one full
    // 128B cache line per run.
    for (int idx = tid; idx < RUNS * 8; idx += 256) {
        const int run = idx >> 3;
        const int q   = idx & 7;
        const int lm  = run >> 4;
        const int nl  = run & 15;
        const int n   = nt * 16 + nl;
        if (n < NKER) {
            const float* s = &lds[run * RSTR + (run >> 7) + q * 4]; // (run>>7)==hi at store time
            float4 val = make_float4(s[0], s[1], s[2], s[3]);
            float* dst = out + (((mt * 16 + lm) * NKER + n) * (OW * OW) + oh * OW + q * 4);
            *(float4*)dst = val;
        }
    }
}

extern "C" void kernel_launch(void* const* d_in, const int* in_sizes, int n_in,
                              void* d_out, int out_size, void* d_ws, size_t ws_size,
                              hipStream_t stream) {
    const float* u   = (const float*)d_in[0];   // (8,32,64,64) f32
    const float* rfs = (const float*)d_in[1];   // (36,64,64)   f32
    float* out = (float*)d_out;                 // (8,32,36,32,32) f32

    const int blocks = 16 * OW * 3;             // m-tiles * oh rows * n-tiles = 1536
    rfpool_wmma_16x16x4<<<blocks, 256, 0, stream>>>(u, rfs, out);
}